// ChamferLoss_39127152067060
// MI455X (gfx1250) — compile-verified
//
#include <hip/hip_runtime.h>
#include <math.h>

typedef float v2f __attribute__((ext_vector_type(2)));
typedef float v8f __attribute__((ext_vector_type(8)));

#define TILE 16
#define QT 2                        // query tiles per wave (A-fragments resident)
#define SEG 8                       // database split into SEG segments (parallelism)
#define CHUNK 1024                  // database points per segment chunk (16 KB LDS)
#define WAVES_PER_BLOCK 8
#define BLOCK (WAVES_PER_BLOCK * 32)

// Build augmented vectors:
//   aug1(p) = (px, py, pz, 1)
//   aug2(p) = (-2px, -2py, -2pz, |p|^2)
//   sq(p)   = |p|^2
__global__ void chamfer_prep(const float* __restrict__ pts,
                             float4* __restrict__ aug1,
                             float4* __restrict__ aug2,
                             float* __restrict__ sq, int total) {
    int i = blockIdx.x * blockDim.x + threadIdx.x;
    if (i >= total) return;
    float px = pts[3 * i + 0];
    float py = pts[3 * i + 1];
    float pz = pts[3 * i + 2];
    float s = px * px + py * py + pz * pz;
    aug1[i] = make_float4(px, py, pz, 1.0f);
    aug2[i] = make_float4(-2.0f * px, -2.0f * py, -2.0f * pz, s);
    sq[i] = s;
}

// seg_min[seg][b][m] = min_{n in segment} ( aug1_q[m] . aug2_db[n] )
//                    = min_{n in segment} ( |r_n|^2 - 2 q_m . r_n )        (|q|^2 added later)
// grid = (Nq / (QT*TILE*WAVES_PER_BLOCK), SEG, B); one wave owns QT 16-row query tiles.
__global__ __launch_bounds__(BLOCK) void chamfer_min_pass(
    const float* __restrict__ q_aug1,   // [B * Nq * 4]
    const float* __restrict__ db_aug2,  // [B * Nd * 4]
    float* __restrict__ seg_min,        // [SEG * B * Nq]
    int Nq, int Nd) {
    __shared__ float s_db[CHUNK * 4];   // 16 KB

    const int b     = blockIdx.z;
    const int seg   = blockIdx.y;
    const int wave  = threadIdx.x >> 5;
    const int lane  = threadIdx.x & 31;
    const int half  = lane >> 4;        // 0: K={0,1}, C rows 0..7 ; 1: K={2,3}, C rows 8..15
    const int col   = lane & 15;
    const int k0    = half * 2;
    const int qbase = (blockIdx.x * WAVES_PER_BLOCK + wave) * (QT * TILE);

    // Stage this segment's chunk: CHUNK float4 across BLOCK threads (4 each)
    const float4* db = (const float4*)db_aug2 + (size_t)b * Nd + (size_t)seg * CHUNK;
#pragma unroll
    for (int t = threadIdx.x; t < CHUNK; t += BLOCK)
        ((float4*)s_db)[t] = db[t];

    // A fragments (16x4 f32): lane holds row M = col, components K = k0, k0+1
    const float* qa = q_aug1 + ((size_t)b * Nq + (qbase + col)) * 4 + k0;
    v2f a0 = *(const v2f*)qa;
    v2f a1 = *(const v2f*)(qa + TILE * 4);

    __syncthreads();

    v8f m0, m1;
#pragma unroll
    for (int r = 0; r < 8; ++r) { m0[r] = 3.0e38f; m1[r] = 3.0e38f; }

#pragma unroll 2
    for (int tile = 0; tile < CHUNK; tile += TILE) {
        // B fragment (4x16 f32): lane holds column N = col, rows K = k0, k0+1
        v2f bfrag = *(const v2f*)&s_db[(tile + col) * 4 + k0];
        v8f z = {0.f, 0.f, 0.f, 0.f, 0.f, 0.f, 0.f, 0.f};
        v8f c0 = __builtin_amdgcn_wmma_f32_16x16x4_f32(
            false, a0, false, bfrag, (short)0, z, false, false);
        v8f c1 = __builtin_amdgcn_wmma_f32_16x16x4_f32(
            false, a1, false, bfrag, (short)0, z, false, false);
#pragma unroll
        for (int r = 0; r < 8; ++r) {
            m0[r] = fminf(m0[r], c0[r]);
            m1[r] = fminf(m1[r], c1[r]);
        }
    }

    // Row-min across 16 columns: xor-shuffle masks <= 8 stay inside each 16-lane half.
#pragma unroll
    for (int r = 0; r < 8; ++r) {
        float v0 = m0[r], v1 = m1[r];
        v0 = fminf(v0, __shfl_xor(v0, 1, 32));
        v1 = fminf(v1, __shfl_xor(v1, 1, 32));
        v0 = fminf(v0, __shfl_xor(v0, 2, 32));
        v1 = fminf(v1, __shfl_xor(v1, 2, 32));
        v0 = fminf(v0, __shfl_xor(v0, 4, 32));
        v1 = fminf(v1, __shfl_xor(v1, 4, 32));
        v0 = fminf(v0, __shfl_xor(v0, 8, 32));
        v1 = fminf(v1, __shfl_xor(v1, 8, 32));
        m0[r] = v0;
        m1[r] = v1;
    }

    // C layout: VGPR r holds row r (lanes 0-15) and row r+8 (lanes 16-31).
    if (col == 0) {
        float* outp = seg_min + ((size_t)seg * gridDim.z + b) * Nq;
#pragma unroll
        for (int r = 0; r < 8; ++r) {
            int row = qbase + r + half * 8;
            outp[row] = m0[r];
            outp[row + TILE] = m1[r];
        }
    }
}

// out = mean_i( min_seg(segx) + xsq ) + mean_j( min_seg(segy) + ysq )
__global__ void chamfer_reduce(const float* __restrict__ segx,
                               const float* __restrict__ xsq,
                               const float* __restrict__ segy,
                               const float* __restrict__ ysq,
                               int nx, int ny, float* __restrict__ out) {
    __shared__ float s[256];
    float accx = 0.f, accy = 0.f;
    for (int i = threadIdx.x; i < nx; i += 256) {
        float v = segx[i];
#pragma unroll
        for (int sg = 1; sg < SEG; ++sg) v = fminf(v, segx[(size_t)sg * nx + i]);
        accx += v + xsq[i];
    }
    for (int i = threadIdx.x; i < ny; i += 256) {
        float v = segy[i];
#pragma unroll
        for (int sg = 1; sg < SEG; ++sg) v = fminf(v, segy[(size_t)sg * ny + i]);
        accy += v + ysq[i];
    }
    s[threadIdx.x] = accx / (float)nx + accy / (float)ny;
    __syncthreads();
    for (int off = 128; off > 0; off >>= 1) {
        if ((int)threadIdx.x < off) s[threadIdx.x] += s[threadIdx.x + off];
        __syncthreads();
    }
    if (threadIdx.x == 0) out[0] = s[0];
}

extern "C" void kernel_launch(void* const* d_in, const int* in_sizes, int n_in,
                              void* d_out, int out_size, void* d_ws, size_t ws_size,
                              hipStream_t stream) {
    const float* x = (const float*)d_in[0];  // [B, N, 3]
    const float* y = (const float*)d_in[1];  // [B, M, 3]
    const int B = 4, N = 8192, M = 8192;

    // Workspace layout (floats); total ~4.25 MB
    float* p = (float*)d_ws;
    float* xa1 = p; p += (size_t)B * N * 4;
    float* xa2 = p; p += (size_t)B * N * 4;
    float* xsq = p; p += (size_t)B * N;
    float* ya1 = p; p += (size_t)B * M * 4;
    float* ya2 = p; p += (size_t)B * M * 4;
    float* ysq = p; p += (size_t)B * M;
    float* segx = p; p += (size_t)SEG * B * N;
    float* segy = p; p += (size_t)SEG * B * M;

    chamfer_prep<<<(B * N + 255) / 256, 256, 0, stream>>>(
        x, (float4*)xa1, (float4*)xa2, xsq, B * N);
    chamfer_prep<<<(B * M + 255) / 256, 256, 0, stream>>>(
        y, (float4*)ya1, (float4*)ya2, ysq, B * M);

    // x -> nearest y
    dim3 grid1(N / (QT * TILE * WAVES_PER_BLOCK), SEG, B);   // (32, 8, 4)
    chamfer_min_pass<<<grid1, BLOCK, 0, stream>>>(xa1, ya2, segx, N, M);

    // y -> nearest x
    dim3 grid2(M / (QT * TILE * WAVES_PER_BLOCK), SEG, B);   // (32, 8, 4)
    chamfer_min_pass<<<grid2, BLOCK, 0, stream>>>(ya1, xa2, segy, M, N);

    chamfer_reduce<<<1, 256, 0, stream>>>(segx, xsq, segy, ysq,
                                          B * N, B * M, (float*)d_out);
}